// RNNModel_90726889160766
// MI455X (gfx1250) — compile-verified
//
#include <hip/hip_runtime.h>

// ---------------------------------------------------------------------------
// RNN scan: x_t = x_{t-1} + alpha*(-x + tanh(x)@J^T + u@b_in^T + c_x + s*n)
// T=750, B=1024, H=100.  fp32 WMMA (v_wmma_f32_16x16x4_f32) for the matmul.
// Grid: 64 workgroups (one per 16-row batch tile), 256 threads (8 wave32).
//   waves 0..6 : N-tiles 0..6 (H padded to 112)
//   wave  7    : stages u[t+1], computes readout at t=0 / t=T-1
// J^T fragments persistent in registers (25 x float2 / lane).
// x persistent in registers (8 VGPRs/lane, WMMA C/D layout).
// tanh(x) through double-buffered LDS -> ONE LDS-only barrier per step
// (raw s_wait_dscnt + s_barrier: global state stores drain off-path).
// Noise loads for step t are issued BEFORE the WMMA chain to overlap.
// ---------------------------------------------------------------------------

#define RNN_T 750
#define RNN_B 1024
#define RNN_H 100
#define RNN_HP 112            // 7 tiles of 16
#define RNN_ALPHA 0.1f        // DT/TAU
#define RNN_NSTD 0.1f

typedef __attribute__((ext_vector_type(2))) float v2f;
typedef __attribute__((ext_vector_type(8))) float v8f;

// LDS-only workgroup barrier: order DS ops, but do NOT wait on global
// store/load counters (states stores keep draining in the background).
__device__ __forceinline__ void lds_barrier() {
    asm volatile("s_wait_dscnt 0x0" ::: "memory");
    __builtin_amdgcn_s_barrier();
    asm volatile("" ::: "memory");
}

__global__ __launch_bounds__(256) void rnn_scan_kernel(
    const float* __restrict__ input_seq,  // [T,B,4]
    const float* __restrict__ noise,      // [T,B,H]
    const float* __restrict__ J,          // [H,H]
    const float* __restrict__ b_in,       // [H,4]
    const float* __restrict__ c_x,        // [H]
    const float* __restrict__ wout_w,     // [1,H]
    const float* __restrict__ wout_b,     // [1]
    float* __restrict__ out)              // [B*2] outputs ++ [T*B*H] states
{
    __shared__ __align__(16) float rbuf[2][16 * RNN_HP]; // tanh(x), double buffered
    __shared__ __align__(16) float ubuf[2][16 * 4];      // u tile, double buffered
    __shared__ __align__(16) float wout_s[RNN_HP];

    const int tid  = threadIdx.x;
    const int wave = tid >> 5;     // 0..7
    const int lane = tid & 31;
    const int half = lane >> 4;    // 0/1: selects K-pair per ISA f32 A layout
    const int l16  = lane & 15;
    const int bb   = blockIdx.x * 16;          // batch base row

    // ---- init LDS -------------------------------------------------------
    for (int i = tid; i < 2 * 16 * RNN_HP; i += 256)
        (&rbuf[0][0])[i] = 0.0f;               // tanh(x_{-1}) = tanh(0) = 0
    if (tid < RNN_HP) wout_s[tid] = (tid < RNN_H) ? wout_w[tid] : 0.0f;
    if (tid < 16) {                            // stage u[t=0]
        const float4 u0 = *(const float4*)(input_seq + ((size_t)bb + tid) * 4);
        *(float4*)&ubuf[0][tid * 4] = u0;
    }

    // ---- per-wave persistent state (waves 0..6) -------------------------
    v2f   jreg[25];                 // J^T fragment for this N-tile, resident
    float xr[8];                    // x tile in WMMA C/D layout
    float bi0 = 0.f, bi1 = 0.f, bi2 = 0.f, bi3 = 0.f, cxn = 0.f;
    const int  ncol   = wave * 16 + l16;              // padded column 0..111
    const bool nvalid = (wave < 7) && (ncol < RNN_H);

    // Persistent streaming pointers (advance by B*H per step).
    // Row owned at unroll index v is m = v + 8*half  ->  offset (v + 8*half)*H.
    const float* nzp = noise + (size_t)(bb + 8 * half) * RNN_H + ncol;
    float*       sp  = out + 2 * RNN_B + (size_t)(bb + 8 * half) * RNN_H + ncol;

    if (wave < 7) {
        #pragma unroll
        for (int k = 0; k < 25; ++k) {
            if (nvalid) {
                // B[4k+2h+j][n] = J[n][4k+2h+j]  (mirror of A's K-pair layout)
                jreg[k] = *(const v2f*)(J + (size_t)ncol * RNN_H + 4 * k + half * 2);
            } else {
                jreg[k] = (v2f){0.f, 0.f};
            }
        }
        if (nvalid) {
            const float4 bi = *(const float4*)(b_in + (size_t)ncol * 4);
            bi0 = bi.x; bi1 = bi.y; bi2 = bi.z; bi3 = bi.w;
            cxn = c_x[ncol];
        }
        #pragma unroll
        for (int v = 0; v < 8; ++v) xr[v] = 0.f;
    }
    const float wb = wout_b[0];
    float* const outputs = out;               // [B,2]

    __syncthreads();                          // full fence once, pre-loop

    // ---- time loop: one LDS barrier per step ----------------------------
    for (int t = 0; t < RNN_T; ++t) {
        const int p = t & 1;   // rbuf[p] = tanh(x_{t-1}); write tanh(x_t) -> p^1
        if (wave < 7) {
            // (1) issue this step's noise loads early: overlap with WMMA chain
            float nz[8];
            if (nvalid) {
                #pragma unroll
                for (int v = 0; v < 8; ++v) nz[v] = nzp[(size_t)v * RNN_H];
            }
            // (2) input term u @ b_in^T for the 8 owned rows (LDS float4 reads)
            const float* up = &ubuf[p][0];
            float itv[8];
            #pragma unroll
            for (int v = 0; v < 8; ++v) {
                const int m = v + half * 8;
                const float4 u = *(const float4*)&up[m * 4];
                itv[v] = u.x * bi0 + u.y * bi1 + u.z * bi2 + u.w * bi3;
            }
            // (3) D = r @ J^T tile : 25 chained K=4 fp32 WMMAs
            v8f c = {};
            const float* rb = &rbuf[p][0];
            #pragma unroll
            for (int k = 0; k < 25; ++k) {
                // A (16x4 f32): lane m=l16, components = K {4k+2h, 4k+2h+1}
                v2f a = *(const v2f*)&rb[l16 * RNN_HP + 4 * k + half * 2];
                c = __builtin_amdgcn_wmma_f32_16x16x4_f32(
                        false, a, false, jreg[k], (short)0, c, false, false);
            }
            // (4) state update + feed tanh(x_t) into the other r buffer
            float* rw = &rbuf[p ^ 1][0];
            #pragma unroll
            for (int v = 0; v < 8; ++v) {
                const int m = v + half * 8;           // C/D layout: M = v + 8*half
                const float x  = xr[v];
                const float xn = nvalid
                    ? x + RNN_ALPHA * (-x + c[v] + itv[v] + cxn + RNN_NSTD * nz[v])
                    : 0.f;                             // padded cols stay 0
                xr[v] = xn;
                rw[m * RNN_HP + ncol] = tanhf(xn);
            }
            // (5) single guarded region: stream states out + prefetch next noise
            if (nvalid) {
                #pragma unroll
                for (int v = 0; v < 8; ++v) {
                    sp[(size_t)v * RNN_H] = xr[v];
                    if (t + 1 < RNN_T)
                        __builtin_prefetch(&nzp[(size_t)v * RNN_H +
                                                (size_t)RNN_B * RNN_H], 0, 1);
                }
            }
            nzp += (size_t)RNN_B * RNN_H;
            sp  += (size_t)RNN_B * RNN_H;
        } else {
            // wave 7: stage u[t+1] into the other buffer
            if (t + 1 < RNN_T && lane < 16) {
                const float4 u = *(const float4*)(input_seq +
                                  ((size_t)(t + 1) * RNN_B + bb + lane) * 4);
                *(float4*)&ubuf[p ^ 1][lane * 4] = u;
            }
            // after iter 0's barrier, rbuf[1] == tanh(states[0]) -> out_first
            if (t == 1 && lane < 16) {
                float s = wb;
                for (int h = 0; h < RNN_H; ++h)
                    s += rbuf[1][lane * RNN_HP + h] * wout_s[h];
                outputs[(size_t)(bb + lane) * 2 + 0] = s;
            }
        }
        lds_barrier();
    }

    // rbuf[T&1] == tanh(states[T-1]) -> out_final
    if (wave == 7 && lane < 16) {
        float s = wb;
        for (int h = 0; h < RNN_H; ++h)
            s += rbuf[RNN_T & 1][lane * RNN_HP + h] * wout_s[h];
        outputs[(size_t)(bb + lane) * 2 + 1] = s;
    }
}

extern "C" void kernel_launch(void* const* d_in, const int* in_sizes, int n_in,
                              void* d_out, int out_size, void* d_ws, size_t ws_size,
                              hipStream_t stream) {
    (void)in_sizes; (void)n_in; (void)out_size; (void)d_ws; (void)ws_size;
    const float* input_seq = (const float*)d_in[0];
    const float* noise     = (const float*)d_in[1];
    const float* J_w       = (const float*)d_in[2];
    const float* b_in      = (const float*)d_in[3];
    const float* c_x       = (const float*)d_in[4];
    const float* wout_w    = (const float*)d_in[5];
    const float* wout_b    = (const float*)d_in[6];
    float* out = (float*)d_out;

    dim3 grid(RNN_B / 16);   // 64 independent batch tiles, no inter-block sync
    dim3 block(256);         // 8 wave32
    rnn_scan_kernel<<<grid, block, 0, stream>>>(
        input_seq, noise, J_w, b_in, c_x, wout_w, wout_b, out);
}